// DHT_Layer_4758823764586
// MI455X (gfx1250) — compile-verified
//
#include <hip/hip_runtime.h>

// Problem constants (match reference)
#define N_IMG 16
#define CIN   256
#define DIM   32
#define H_IN  128
#define W_IN  128
#define P_IN  (H_IN * W_IN)     // 16384
#define NA    192               // numangle
#define NR    192               // numrho
#define HW2   (NA * NR)         // 36864
#define BN_EPS 1e-5f
#define KT    16                // conv1 K-tile (channels per staged LDS tile)

typedef __attribute__((ext_vector_type(2))) float v2f;
typedef __attribute__((ext_vector_type(8))) float v8f;
typedef __attribute__((ext_vector_type(4))) int   v4i_t;

__device__ __forceinline__ v8f wmma_f32(v2f a, v2f b, v8f c) {
  // V_WMMA_F32_16X16X4_F32 : D(16x16,f32) = A(16x4,f32) * B(4x16,f32) + C
  return __builtin_amdgcn_wmma_f32_16x16x4_f32(
      /*neg_a=*/false, a, /*neg_b=*/false, b,
      /*c_mod=*/(short)0, c, /*reuse_a=*/false, /*reuse_b=*/false);
}

// --- async global->LDS staging (GLOBAL_LOAD_ASYNC_TO_LDS_B128, ASYNCcnt) ---
#define AS_GLOBAL __attribute__((address_space(1)))
#define AS_LDS    __attribute__((address_space(3)))

#if __has_builtin(__builtin_amdgcn_global_load_async_to_lds_b128) && \
    __has_builtin(__builtin_amdgcn_s_wait_asynccnt)
#define HAVE_ASYNC_LDS 1
__device__ __forceinline__ void async_cp16(const float* g, float* l) {
  // prototype: (v4i addrspace(1)*, v4i addrspace(3)*, imm offset, imm cpol)
  __builtin_amdgcn_global_load_async_to_lds_b128(
      (AS_GLOBAL v4i_t*)g, (AS_LDS v4i_t*)l, 0, 0);
}
#define ASYNC_WAIT(n) __builtin_amdgcn_s_wait_asynccnt(n)
#else
#define HAVE_ASYNC_LDS 0
__device__ __forceinline__ void async_cp16(const float* g, float* l) {
  *(float4*)l = *(const float4*)g;   // fallback: b128 load + ds store
}
#define ASYNC_WAIT(n) ((void)0)
#endif

// ---------------------------------------------------------------------------
// zero small stats buffer (must re-zero every call: atomics accumulate)
// ---------------------------------------------------------------------------
__global__ void k_zero(float* __restrict__ p, int n) {
  int i = blockIdx.x * 256 + threadIdx.x;
  if (i < n) p[i] = 0.f;
}

// ---------------------------------------------------------------------------
// conv1: 1x1 conv as WMMA GEMM with double-buffered async LDS staging.
// grid = N_IMG * H_IN blocks (one image row = 128 px), 8 waves/block,
// wave w owns pixels [w*16, +16) of the row, all 32 out channels.
// K pipeline: 16 tiles of 16 channels; tile = 16ch x 128px = 8 KB staged via
// block-wide GLOBAL_LOAD_ASYNC_TO_LDS_B128 (2 instr/thread/tile).
// ---------------------------------------------------------------------------
__global__ __launch_bounds__(256)
void k_conv1(const float* __restrict__ x, const float* __restrict__ w1,
             const float* __restrict__ bias, float* __restrict__ h1,
             float* __restrict__ gsum, float* __restrict__ gssum) {
  __shared__ float wlds[CIN * DIM];       // [k][m]: lane==m -> conflict-free
  __shared__ float xt[2][KT * W_IN];      // double-buffered activation tiles
  __shared__ float s_sum[DIM], s_ssum[DIM];
  const int tid = threadIdx.x, lane = tid & 31, wid = tid >> 5;

  // weights: w1 is [m][k] (DIM x CIN); store transposed [k][m]
  for (int i = tid; i < DIM * CIN; i += 256) {
    int m = i >> 8, k = i & 255;
    wlds[k * DIM + m] = w1[i];
  }
  if (tid < DIM) { s_sum[tid] = 0.f; s_ssum[tid] = 0.f; }

  const int n = blockIdx.x >> 7;          // image
  const int y = blockIdx.x & 127;         // row
  const float* xrow = x + (size_t)n * CIN * P_IN + y * W_IN;

  // staging geometry: thread copies 2x16B; op0 -> channel wid, op1 -> wid+8
  const int spx = (tid & 31) * 4;
  auto stage = [&](int tile, int buf) {
    const int kb = tile * KT;
    async_cp16(xrow + (size_t)(kb + wid)     * P_IN + spx, &xt[buf][ wid      * W_IN + spx]);
    async_cp16(xrow + (size_t)(kb + wid + 8) * P_IN + spx, &xt[buf][(wid + 8) * W_IN + spx]);
  };
  stage(0, 0);
  stage(1, 1);

  const int khalf = (lane >> 4) * 2;      // k sub-offset of this half-wave
  const int m0  = lane & 15;
  const int px16 = wid * 16 + (lane & 15);
  v8f acc0 = {}; v8f acc1 = {};

  for (int t = 0; t < CIN / KT; ++t) {
    const int buf = t & 1;
    if (t < CIN / KT - 2) ASYNC_WAIT(2);  // tile t's 2 ops done (t+1 in flight)
    else                  ASYNC_WAIT(0);  // drain at pipeline tail
    __syncthreads();                      // staged data visible to all waves
    const int kb = t * KT;
    #pragma unroll
    for (int k4 = 0; k4 < KT; k4 += 4) {
      const int kl = k4 + khalf;          // local k held by this half-wave
      const int ka = kb + kl;             // absolute k (weight index)
      v2f bb, a0, a1;
      bb.x = xt[buf][kl * W_IN + px16];   // B(k, px)
      bb.y = xt[buf][(kl + 1) * W_IN + px16];
      a0.x = wlds[ka * DIM + m0];         // A(m, k), m = 0..15
      a0.y = wlds[(ka + 1) * DIM + m0];
      a1.x = wlds[ka * DIM + m0 + 16];    // A(m, k), m = 16..31
      a1.y = wlds[(ka + 1) * DIM + m0 + 16];
      acc0 = wmma_f32(a0, bb, acc0);
      acc1 = wmma_f32(a1, bb, acc1);
    }
    __syncthreads();                      // all waves done reading buf
    if (t + 2 < CIN / KT) stage(t + 2, buf);
  }

  // D layout: element (m = r + 8*(lane>>4), n = lane&15) in vgpr r
  const int mrow = 8 * (lane >> 4);
  float* hb = h1 + (size_t)n * DIM * P_IN + y * W_IN + px16;
  #pragma unroll
  for (int r = 0; r < 8; ++r) {
    const int m = r + mrow;
    float v0 = acc0[r] + bias[m];
    float v1 = acc1[r] + bias[m + 16];
    hb[(size_t)m * P_IN]        = v0;
    hb[(size_t)(m + 16) * P_IN] = v1;
    atomicAdd(&s_sum[m], v0);       atomicAdd(&s_ssum[m], v0 * v0);
    atomicAdd(&s_sum[m + 16], v1);  atomicAdd(&s_ssum[m + 16], v1 * v1);
  }
  __syncthreads();
  if (tid < DIM) {
    unsafeAtomicAdd(&gsum[tid],  s_sum[tid]);
    unsafeAtomicAdd(&gssum[tid], s_ssum[tid]);
  }
}

// ---------------------------------------------------------------------------
// BN finalize: scale = g*rsqrt(var+eps), shift = beta - mean*scale
// ---------------------------------------------------------------------------
__global__ void k_bnfin(const float* __restrict__ sum, const float* __restrict__ ssum,
                        const float* __restrict__ g, const float* __restrict__ be,
                        float* __restrict__ scale, float* __restrict__ shift,
                        float inv_count) {
  int c = threadIdx.x;
  if (c < DIM) {
    float m = sum[c] * inv_count;
    float v = ssum[c] * inv_count - m * m;
    float s = g[c] * rsqrtf(v + BN_EPS);
    scale[c] = s;
    shift[c] = be[c] - m * s;
  }
}

// ---------------------------------------------------------------------------
// DHT: one block per (n, c) plane.  147 KB LDS accumulator (of 320 KB WGP
// LDS); BN1+ReLU folded into the single global read of each input element.
// Pixel-outer / angle-inner: each plane element read from HBM exactly once,
// all accumulation through ds_add_f32.
// ---------------------------------------------------------------------------
__global__ __launch_bounds__(256)
void k_dht(const float* __restrict__ h1, const float* __restrict__ scale,
           const float* __restrict__ shift, float* __restrict__ acc) {
  extern __shared__ float smem[];
  float* accL = smem;            // HW2 floats
  float* tabS = smem + HW2;      // NA
  float* tabC = tabS + NA;       // NA
  const int tid = threadIdx.x;
  const int n = blockIdx.x >> 5;
  const int c = blockIdx.x & 31;

  for (int i = tid; i < HW2; i += 256) accL[i] = 0.f;
  if (tid < NA) {
    const float inv_irho = 191.0f / 182.0f;   // irho = int(sqrt(2*128^2)+1)/(NR-1)
    float th = (float)tid * (3.14159265358979323846f / (float)NA);
    tabS[tid] = __sinf(th) * inv_irho;
    tabC[tid] = __cosf(th) * inv_irho;
  }
  __syncthreads();

  const float sc = scale[c], sh = shift[c];
  const float* plane = h1 + ((size_t)n * DIM + c) * P_IN;
  for (int p = tid; p < P_IN; p += 256) {
    float v = fmaxf(plane[p] * sc + sh, 0.f);   // BN1 + ReLU
    if (v != 0.f) {                             // ReLU sparsity: skip zeros
      float xc = (float)(p & 127) - 64.f;
      float yc = (float)(p >> 7) - 64.f;
      #pragma unroll 4
      for (int a = 0; a < NA; ++a) {
        int r = __float2int_rn(xc * tabC[a] + yc * tabS[a]) + NR / 2;
        r = min(max(r, 0), NR - 1);
        atomicAdd(&accL[a * NR + r], v);        // ds_add_f32
      }
    }
  }
  __syncthreads();
  float* ob = acc + ((size_t)n * DIM + c) * HW2;
  for (int i = tid; i < HW2; i += 256) ob[i] = accL[i];
}

// ---------------------------------------------------------------------------
// 3x3 conv (used for conv2 and conv3) as WMMA GEMM with K = 32*9 = 288.
// grid = N_IMG * (12 x-tiles * 24 y-tiles); block covers 8 rows x 16 cols,
// wave w owns output row y0+w.  Optional folded per-channel input transform
// (BN+ReLU of the previous layer) applied during the halo-patch load.
// ---------------------------------------------------------------------------
__global__ __launch_bounds__(256)
void k_conv3x3(const float* __restrict__ in, const float* __restrict__ w,
               const float* __restrict__ bias,
               const float* __restrict__ inScale, const float* __restrict__ inShift,
               float* __restrict__ out,
               float* __restrict__ gsum, float* __restrict__ gssum) {
  __shared__ float wlds[DIM * 289];          // 32 x 288, stride 289 (odd: conflict-free)
  __shared__ float patch[DIM * 10 * 18];     // 32 ic x (8+2) x (16+2) halo tile
  __shared__ float s_sum[DIM], s_ssum[DIM];
  const int tid = threadIdx.x, lane = tid & 31, wid = tid >> 5;

  for (int i = tid; i < DIM * 288; i += 256) {
    int m = i / 288, k = i - m * 288;        // k = ic*9 + dy*3 + dx (OIHW flat)
    wlds[m * 289 + k] = w[i];
  }
  if (tid < DIM) { s_sum[tid] = 0.f; s_ssum[tid] = 0.f; }

  const int b  = blockIdx.x;
  const int n  = b / 288;
  const int t  = b - n * 288;
  const int ty = t / 12, tx = t - ty * 12;
  const int x0 = tx * 16, y0 = ty * 8;
  const float* ib = in + (size_t)n * DIM * HW2;
  const bool xform = (inScale != nullptr);

  for (int i = tid; i < DIM * 180; i += 256) {
    int ic  = i / 180;
    int rem = i - ic * 180;
    int ry = rem / 18, rx = rem - ry * 18;
    int gy = y0 + ry - 1, gx = x0 + rx - 1;
    float v = 0.f;
    if (gy >= 0 && gy < NR && gx >= 0 && gx < NR) {
      v = ib[(size_t)ic * HW2 + gy * NR + gx];
      if (xform) v = fmaxf(v * inScale[ic] + inShift[ic], 0.f);  // BN+ReLU fold
    }
    patch[i] = v;
  }
  __syncthreads();

  const int khalf = (lane >> 4) * 2;
  const int m0  = lane & 15;
  const int col = lane & 15;
  v8f acc0 = {}; v8f acc1 = {};
  for (int k0 = 0; k0 < 288; k0 += 4) {
    v2f a0, a1, bb;
    #pragma unroll
    for (int v = 0; v < 2; ++v) {
      int k  = k0 + khalf + v;
      int ic = k / 9;
      int tt = k - ic * 9;
      int dy = tt / 3;
      int dx = tt - dy * 3;
      float bv  = patch[ic * 180 + (wid + dy) * 18 + col + dx];
      float av0 = wlds[m0 * 289 + k];
      float av1 = wlds[(m0 + 16) * 289 + k];
      if (v == 0) { bb.x = bv; a0.x = av0; a1.x = av1; }
      else        { bb.y = bv; a0.y = av0; a1.y = av1; }
    }
    acc0 = wmma_f32(a0, bb, acc0);
    acc1 = wmma_f32(a1, bb, acc1);
  }

  const int oy = y0 + wid, ox = x0 + col;
  const int mrow = 8 * (lane >> 4);
  float* ob = out + (size_t)n * DIM * HW2 + oy * NR + ox;
  #pragma unroll
  for (int r = 0; r < 8; ++r) {
    const int m = r + mrow;
    float v0 = acc0[r] + bias[m];
    float v1 = acc1[r] + bias[m + 16];
    ob[(size_t)m * HW2]        = v0;
    ob[(size_t)(m + 16) * HW2] = v1;
    atomicAdd(&s_sum[m], v0);       atomicAdd(&s_ssum[m], v0 * v0);
    atomicAdd(&s_sum[m + 16], v1);  atomicAdd(&s_ssum[m + 16], v1 * v1);
  }
  __syncthreads();
  if (tid < DIM) {
    unsafeAtomicAdd(&gsum[tid],  s_sum[tid]);
    unsafeAtomicAdd(&gssum[tid], s_ssum[tid]);
  }
}

// ---------------------------------------------------------------------------
// final BN3 + ReLU, in place on d_out
// ---------------------------------------------------------------------------
__global__ void k_bnapply(float* __restrict__ io, const float* __restrict__ scale,
                          const float* __restrict__ shift) {
  size_t i = (size_t)blockIdx.x * 256 + threadIdx.x;
  int c = (int)((i / HW2) & 31);
  io[i] = fmaxf(io[i] * scale[c] + shift[c], 0.f);
}

// ---------------------------------------------------------------------------
extern "C" void kernel_launch(void* const* d_in, const int* in_sizes, int n_in,
                              void* d_out, int out_size, void* d_ws, size_t ws_size,
                              hipStream_t stream) {
  (void)in_sizes; (void)n_in; (void)out_size; (void)ws_size;
  const float* x   = (const float*)d_in[0];
  const float* w1  = (const float*)d_in[1];
  const float* b1  = (const float*)d_in[2];
  const float* g1  = (const float*)d_in[3];
  const float* be1 = (const float*)d_in[4];
  const float* w2  = (const float*)d_in[5];
  const float* b2  = (const float*)d_in[6];
  const float* g2  = (const float*)d_in[7];
  const float* be2 = (const float*)d_in[8];
  const float* w3  = (const float*)d_in[9];
  const float* b3  = (const float*)d_in[10];
  const float* g3  = (const float*)d_in[11];
  const float* be3 = (const float*)d_in[12];

  float* ws  = (float*)d_ws;
  float* h1  = ws;                                   // 16*32*16384
  float* acc = h1 + (size_t)N_IMG * DIM * P_IN;      // 16*32*36864
  float* h2  = acc + (size_t)N_IMG * DIM * HW2;      // 16*32*36864
  float* st  = h2 + (size_t)N_IMG * DIM * HW2;       // stats (6*32) + scale/shift (6*32)
  float *sum0 = st,       *ss0 = st + 32;
  float *sum1 = st + 64,  *ss1 = st + 96;
  float *sum2 = st + 128, *ss2 = st + 160;
  float* sc = st + 192;
  float *scale0 = sc,       *shift0 = sc + 32;
  float *scale1 = sc + 64,  *shift1 = sc + 96;
  float *scale2 = sc + 128, *shift2 = sc + 160;
  float* outp = (float*)d_out;

  k_zero<<<1, 256, 0, stream>>>(st, 192);

  // layer 1: 1x1 conv (WMMA + async LDS staging) + batch stats
  k_conv1<<<N_IMG * H_IN, 256, 0, stream>>>(x, w1, b1, h1, sum0, ss0);
  k_bnfin<<<1, 32, 0, stream>>>(sum0, ss0, g1, be1, scale0, shift0,
                                1.0f / ((float)N_IMG * (float)P_IN));

  // deep Hough transform with BN1+ReLU folded into the read
  size_t dht_lds = (size_t)(HW2 + 2 * NA) * sizeof(float);
  k_dht<<<N_IMG * DIM, 256, dht_lds, stream>>>(h1, scale0, shift0, acc);

  // layer 2: 3x3 conv (WMMA) on raw accumulator + stats
  k_conv3x3<<<N_IMG * 288, 256, 0, stream>>>(acc, w2, b2, nullptr, nullptr,
                                             h2, sum1, ss1);
  k_bnfin<<<1, 32, 0, stream>>>(sum1, ss1, g2, be2, scale1, shift1,
                                1.0f / ((float)N_IMG * (float)HW2));

  // layer 3: 3x3 conv (WMMA) with BN2+ReLU folded into patch load; raw -> d_out
  k_conv3x3<<<N_IMG * 288, 256, 0, stream>>>(h2, w3, b3, scale1, shift1,
                                             outp, sum2, ss2);
  k_bnfin<<<1, 32, 0, stream>>>(sum2, ss2, g3, be3, scale2, shift2,
                                1.0f / ((float)N_IMG * (float)HW2));

  // BN3 + ReLU in place
  k_bnapply<<<(N_IMG * DIM * HW2) / 256, 256, 0, stream>>>(outp, scale2, shift2);
}